// LeNetDeform_44195213476585
// MI455X (gfx1250) — compile-verified
//
#include <hip/hip_runtime.h>
#include <hip/hip_bf16.h>

typedef __attribute__((ext_vector_type(2))) float v2f;
typedef __attribute__((ext_vector_type(8))) float v8f;

// ---------- WMMA helper: exact fp32 D = A(16x4) * B(4x16) + C ----------
__device__ __forceinline__ v8f wmma4(v2f a, v2f b, v8f c) {
  return __builtin_amdgcn_wmma_f32_16x16x4_f32(
      /*neg_a=*/false, a, /*neg_b=*/false, b,
      /*c_mod=*/(short)0, c, /*reuse_a=*/false, /*reuse_b=*/false);
}

// ---------- conv1 (3x3, 3->6) + ReLU, scalar ----------
__global__ void conv1_relu_k(const float* __restrict__ x,
                             const float* __restrict__ w,
                             const float* __restrict__ b,
                             float* __restrict__ y, int total) {
  int i = blockIdx.x * blockDim.x + threadIdx.x;
  if (i >= total) return;                       // total = N*6*30*30
  int xw = i % 30; int t = i / 30;
  int yh = t % 30; t /= 30;
  int o  = t % 6;  int n = t / 6;
  const float* xp = x + (size_t)n * 3 * 32 * 32;
  float acc = b[o];
#pragma unroll
  for (int c = 0; c < 3; ++c)
#pragma unroll
    for (int ky = 0; ky < 3; ++ky)
#pragma unroll
      for (int kx = 0; kx < 3; ++kx)
        acc += xp[(c * 32 + yh + ky) * 32 + xw + kx] * w[((o * 3 + c) * 3 + ky) * 3 + kx];
  y[i] = fmaxf(acc, 0.f);
}

// ---------- fused: dconv1(3x3,6->6) -> dconv2(1x1,6->108) -> deform conv -> ReLU ----------
__global__ void deform_fused_k(const float* __restrict__ h1,
                               const float* __restrict__ d1w, const float* __restrict__ d1b,
                               const float* __restrict__ d2w, const float* __restrict__ d2b,
                               const float* __restrict__ dw,  const float* __restrict__ db,
                               float* __restrict__ h2, int total) {
  int i = blockIdx.x * blockDim.x + threadIdx.x;
  if (i >= total) return;                       // total = N*28*28
  int wo = i % 28; int t = i / 28;
  int ho = t % 28; int n = t / 28;
  const float* hp = h1 + (size_t)n * 6 * 900;   // h1: [N,6,30,30]

  // 6x3x3 patch of h1 at (ho..ho+2, wo..wo+2)
  float patch[54];
#pragma unroll
  for (int c = 0; c < 6; ++c)
#pragma unroll
    for (int ky = 0; ky < 3; ++ky)
#pragma unroll
      for (int kx = 0; kx < 3; ++kx)
        patch[(c * 3 + ky) * 3 + kx] = hp[c * 900 + (ho + ky) * 30 + (wo + kx)];

  // d1 = dconv1 output pixel (6 channels)
  float d1[6];
#pragma unroll
  for (int co = 0; co < 6; ++co) {
    float a = d1b[co];
#pragma unroll
    for (int j = 0; j < 54; ++j) a += d1w[co * 54 + j] * patch[j];
    d1[co] = a;
  }

  float acc[6];
#pragma unroll
  for (int o = 0; o < 6; ++o) acc[o] = db[o];

  for (int c = 0; c < 6; ++c) {
    const float* hc = hp + c * 900;
#pragma unroll
    for (int kk = 0; kk < 9; ++kk) {
      int jy = (c * 9 + kk) * 2;                // dconv2 out channel for dy; jy+1 for dx
      float dy = d2b[jy], dx = d2b[jy + 1];
#pragma unroll
      for (int q = 0; q < 6; ++q) {
        dy += d2w[jy * 6 + q] * d1[q];
        dx += d2w[(jy + 1) * 6 + q] * d1[q];
      }
      float ys = dy + (float)(kk / 3) + (float)ho;
      float xs = dx + (float)(kk % 3) + (float)wo;
      float y0f = floorf(ys), x0f = floorf(xs);
      float wy = ys - y0f, wx = xs - x0f;
      int y0 = (int)y0f, x0 = (int)x0f;
      auto G = [&](int yy, int xx) -> float {
        if (yy < 0 || yy >= 30 || xx < 0 || xx >= 30) return 0.f;
        return hc[yy * 30 + xx];
      };
      float s = (1.f - wy) * (1.f - wx) * G(y0, x0)
              + (1.f - wy) * wx        * G(y0, x0 + 1)
              + wy        * (1.f - wx) * G(y0 + 1, x0)
              + wy        * wx         * G(y0 + 1, x0 + 1);
#pragma unroll
      for (int o = 0; o < 6; ++o) acc[o] += dw[(o * 6 + c) * 9 + kk] * s;
    }
  }
#pragma unroll
  for (int o = 0; o < 6; ++o)
    h2[(((size_t)n * 6 + o) * 28 + ho) * 28 + wo] = fmaxf(acc[o], 0.f);
}

// ---------- 2x2 maxpool ----------
__global__ void maxpool2_k(const float* __restrict__ in, float* __restrict__ out,
                           int C, int H, int W, int total) {
  int i = blockIdx.x * blockDim.x + threadIdx.x;
  if (i >= total) return;                       // total = N*C*(H/2)*(W/2)
  int Wo = W >> 1, Ho = H >> 1;
  int x = i % Wo; int t = i / Wo;
  int y = t % Ho; t /= Ho;
  int c = t % C;  int n = t / C;
  const float* p = in + (((size_t)n * C + c) * H + 2 * y) * W + 2 * x;
  out[i] = fmaxf(fmaxf(p[0], p[1]), fmaxf(p[W], p[W + 1]));
}

// ---------- implicit-GEMM conv (3x3, Cout=16) with WMMA f32 16x16x4, + bias + ReLU ----------
// M = N*Hout*Wout rows; N-dim = Cout = 16 (one tile); K = Cin*9 (padded to mult of 4).
__global__ void conv_wmma_k(const float* __restrict__ in, const float* __restrict__ w,
                            const float* __restrict__ bias, float* __restrict__ out,
                            int Cin, int Hin, int Win, int Hout, int Wout,
                            int K, int Mtotal, int tilesM) {
  int lane = threadIdx.x & 31;
  int wave = threadIdx.x >> 5;
  int tile = blockIdx.x * (blockDim.x >> 5) + wave;
  if (tile >= tilesM) return;                   // wave-uniform: EXEC stays all-ones
  int r  = lane & 15;                           // A row / D col index
  int kh = lane >> 4;                           // K-half select
  int m  = tile * 16 + r;
  bool rowok = m < Mtotal;
  int mm = rowok ? m : 0;
  int xx = mm % Wout; int tt = mm / Wout;
  int yy = tt % Hout; int nn = tt / Hout;
  const float* inp = in + (size_t)nn * Cin * Hin * Win;
  int Kpad = (K + 3) & ~3;

  v8f acc = {};
  for (int k0 = 0; k0 < Kpad; k0 += 4) {
    int k = k0 + 2 * kh;
    v2f a, b;
    float ax = 0.f, ay = 0.f;
    if (rowok && k < K) {
      int ci = k / 9, rem = k % 9;
      ax = inp[(ci * Hin + yy + rem / 3) * Win + xx + rem % 3];
    }
    if (rowok && k + 1 < K) {
      int k1 = k + 1;
      int ci = k1 / 9, rem = k1 % 9;
      ay = inp[(ci * Hin + yy + rem / 3) * Win + xx + rem % 3];
    }
    a.x = ax; a.y = ay;
    b.x = (k     < K) ? w[r * K + k]     : 0.f;   // B[k][col], col = lane&15 (= r)
    b.y = (k + 1 < K) ? w[r * K + k + 1] : 0.f;
    acc = wmma4(a, b, acc);
  }
  float bv = bias[r];
#pragma unroll
  for (int v = 0; v < 8; ++v) {
    int m2 = tile * 16 + v + 8 * kh;            // D: M = v + 8*(lane>=16), N = lane&15
    if (m2 >= Mtotal) continue;
    int x2 = m2 % Wout; int t2 = m2 / Wout;
    int y2 = t2 % Hout; int n2 = t2 / Hout;
    float val = fmaxf(acc[v] + bv, 0.f);
    out[(((size_t)n2 * 16 + r) * Hout + y2) * Wout + x2] = val;
  }
}

// ---------- FC layer: out[M x Nout] = A[M x K] * W[Nout x K]^T + b, WMMA f32 ----------
__global__ void fc_wmma_k(const float* __restrict__ A, const float* __restrict__ W,
                          const float* __restrict__ bias, float* __restrict__ out,
                          int M, int K, int Nout, int relu, int tilesM) {
  int lane = threadIdx.x & 31;
  int wave = threadIdx.x >> 5;
  int tileM = blockIdx.x * (blockDim.x >> 5) + wave;
  if (tileM >= tilesM) return;                  // wave-uniform guard
  int tileN = blockIdx.y;
  int r  = lane & 15;
  int kh = lane >> 4;
  int row = tileM * 16 + r;
  int col = tileN * 16 + r;
  bool rowok = row < M;
  bool colok = col < Nout;
  const float* arow = A + (size_t)(rowok ? row : 0) * K;
  const float* wrow = W + (size_t)(colok ? col : 0) * K;

  v8f acc = {};
  for (int k0 = 0; k0 < K; k0 += 4) {           // K is a multiple of 4 (400/120/84)
    int k = k0 + 2 * kh;
    v2f a, b;
    a.x = rowok ? arow[k]     : 0.f;
    a.y = rowok ? arow[k + 1] : 0.f;
    b.x = colok ? wrow[k]     : 0.f;
    b.y = colok ? wrow[k + 1] : 0.f;
    acc = wmma4(a, b, acc);
  }
  float bv = colok ? bias[col] : 0.f;
#pragma unroll
  for (int v = 0; v < 8; ++v) {
    int rr = tileM * 16 + v + 8 * kh;
    if (!colok || rr >= M) continue;
    float val = acc[v] + bv;
    if (relu) val = fmaxf(val, 0.f);
    out[(size_t)rr * Nout + col] = val;
  }
}

extern "C" void kernel_launch(void* const* d_in, const int* in_sizes, int n_in,
                              void* d_out, int out_size, void* d_ws, size_t ws_size,
                              hipStream_t stream) {
  const float* x        = (const float*)d_in[0];
  const float* conv1_w  = (const float*)d_in[1];
  const float* conv1_b  = (const float*)d_in[2];
  const float* dconv1_w = (const float*)d_in[3];
  const float* dconv1_b = (const float*)d_in[4];
  const float* dconv2_w = (const float*)d_in[5];
  const float* dconv2_b = (const float*)d_in[6];
  const float* dw       = (const float*)d_in[7];
  const float* db       = (const float*)d_in[8];
  const float* conv3_w  = (const float*)d_in[9];
  const float* conv3_b  = (const float*)d_in[10];
  const float* conv4_w  = (const float*)d_in[11];
  const float* conv4_b  = (const float*)d_in[12];
  const float* fc1_w    = (const float*)d_in[13];
  const float* fc1_b    = (const float*)d_in[14];
  const float* fc2_w    = (const float*)d_in[15];
  const float* fc2_b    = (const float*)d_in[16];
  const float* fc3_w    = (const float*)d_in[17];
  const float* fc3_b    = (const float*)d_in[18];
  float* outp = (float*)d_out;

  const int N = in_sizes[0] / (3 * 32 * 32);

  float* ws = (float*)d_ws;
  float* h1 = ws;                                   // [N,6,30,30]
  float* h2 = h1 + (size_t)N * 6 * 30 * 30;         // [N,6,28,28]
  float* p1 = h2 + (size_t)N * 6 * 28 * 28;         // [N,6,14,14]
  float* h3 = p1 + (size_t)N * 6 * 14 * 14;         // [N,16,12,12]
  float* h4 = h3 + (size_t)N * 16 * 12 * 12;        // [N,16,10,10]
  float* p2 = h4 + (size_t)N * 16 * 10 * 10;        // [N,400]
  float* f1 = p2 + (size_t)N * 400;                 // [N,120]
  float* f2 = f1 + (size_t)N * 120;                 // [N,84]

  // conv1 + relu
  {
    int total = N * 6 * 30 * 30;
    conv1_relu_k<<<(total + 255) / 256, 256, 0, stream>>>(x, conv1_w, conv1_b, h1, total);
  }
  // fused offset convs + deformable conv + relu
  {
    int total = N * 28 * 28;
    deform_fused_k<<<(total + 255) / 256, 256, 0, stream>>>(
        h1, dconv1_w, dconv1_b, dconv2_w, dconv2_b, dw, db, h2, total);
  }
  // maxpool 28->14
  {
    int total = N * 6 * 14 * 14;
    maxpool2_k<<<(total + 255) / 256, 256, 0, stream>>>(h2, p1, 6, 28, 28, total);
  }
  // conv3 (6->16, 14->12) + relu, WMMA implicit GEMM
  {
    int M = N * 12 * 12, K = 54;
    int tiles = (M + 15) / 16;
    int blocks = (tiles + 3) / 4;
    conv_wmma_k<<<blocks, 128, 0, stream>>>(p1, conv3_w, conv3_b, h3,
                                            6, 14, 14, 12, 12, K, M, tiles);
  }
  // conv4 (16->16, 12->10) + relu, WMMA implicit GEMM
  {
    int M = N * 10 * 10, K = 144;
    int tiles = (M + 15) / 16;
    int blocks = (tiles + 3) / 4;
    conv_wmma_k<<<blocks, 128, 0, stream>>>(h3, conv4_w, conv4_b, h4,
                                            16, 12, 12, 10, 10, K, M, tiles);
  }
  // maxpool 10->5  (layout [N,16,5,5] == flattened [N,400])
  {
    int total = N * 16 * 5 * 5;
    maxpool2_k<<<(total + 255) / 256, 256, 0, stream>>>(h4, p2, 16, 10, 10, total);
  }
  // fc1 + relu
  {
    int M = N, K = 400, Nout = 120;
    int tiles = (M + 15) / 16;
    dim3 grid((tiles + 3) / 4, (Nout + 15) / 16);
    fc_wmma_k<<<grid, 128, 0, stream>>>(p2, fc1_w, fc1_b, f1, M, K, Nout, 1, tiles);
  }
  // fc2 + relu
  {
    int M = N, K = 120, Nout = 84;
    int tiles = (M + 15) / 16;
    dim3 grid((tiles + 3) / 4, (Nout + 15) / 16);
    fc_wmma_k<<<grid, 128, 0, stream>>>(f1, fc2_w, fc2_b, f2, M, K, Nout, 1, tiles);
  }
  // fc3 (no relu) -> d_out
  {
    int M = N, K = 84, Nout = 10;
    int tiles = (M + 15) / 16;
    dim3 grid((tiles + 3) / 4, (Nout + 15) / 16);
    fc_wmma_k<<<grid, 128, 0, stream>>>(f2, fc3_w, fc3_b, outp, M, K, Nout, 0, tiles);
  }
}